// Fusion_Loss_73916387164734
// MI455X (gfx1250) — compile-verified
//
#include <hip/hip_runtime.h>
#include <stdint.h>

typedef __attribute__((ext_vector_type(2))) float v2f;
typedef __attribute__((ext_vector_type(8))) float v8f;

namespace {
constexpr int KCLS = 64;                 // classes / prototypes (= N of the GEMM)
constexpr int QPC  = 512;                // queries per class
constexpr int D    = 1024;               // feature dim (= K of the GEMM)
constexpr int M    = KCLS * QPC;         // 32768 query rows
constexpr int KC   = 32;                 // K-chunk staged per pipeline stage
constexpr int NCHUNKS = D / KC;          // 32
constexpr int ROWS = 256;                // query rows per workgroup (8 waves x 2 tiles x 16)
constexpr int AST  = KC + 4;             // 36: xq chunk stride (floats), conflict-free
constexpr int BST  = KC + 4;             // 36: proto chunk stride (floats), row-major [n][kk]
constexpr int CST  = 68;                 // C spill stride (floats), conflict-free
constexpr int ABUF = ROWS * AST;         // 9216 floats per A buffer
constexpr int BBUF = KCLS * BST;         // 2304 floats per B buffer
constexpr int OFF_A0 = 0;
constexpr int OFF_A1 = ABUF;
constexpr int OFF_B0 = 2 * ABUF;         // 18432
constexpr int OFF_B1 = OFF_B0 + BBUF;
constexpr int OFF_P  = OFF_B0 + 2 * BBUF;        // 23040
constexpr int SMEM_FLOATS = OFF_P + KCLS;        // 23104 floats = 92416 B
}

// Zero the two output scalars (harness poisons d_out; main kernel atomically adds).
__global__ void zero_out_kernel(float* __restrict__ out) {
  if (threadIdx.x < 2) out[threadIdx.x] = 0.0f;
}

// LDS generic pointers keep the LDS byte address in the low 32 bits (aperture
// addressing: addr[63:32] selects the shared aperture, addr[31:0] is the offset).
__device__ __forceinline__ uint32_t lds_addr_of(const float* p) {
  return (uint32_t)(uintptr_t)p;
}

// CDNA5 async copy: 16 bytes per lane, global -> LDS, tracked by ASYNCcnt.
__device__ __forceinline__ void async_copy_b128(uint32_t lds_byte_addr, const float* g) {
  asm volatile("global_load_async_to_lds_b128 %0, %1, off"
               :: "v"(lds_byte_addr), "v"((unsigned long long)(uintptr_t)g)
               : "memory");
}

__device__ __forceinline__ void wait_async_all() {
  asm volatile("s_wait_asynccnt 0" ::: "memory");
}

__device__ __forceinline__ void issue_chunk_loads(float* smem, int bufsel,
                                                  const float* __restrict__ xq,
                                                  const float* __restrict__ proto,
                                                  long rowBase, int kc, int tid) {
  const uint32_t aBase = lds_addr_of(smem + (bufsel ? OFF_A1 : OFF_A0));
  const uint32_t bBase = lds_addr_of(smem + (bufsel ? OFF_B1 : OFF_B0));
  // A: 256 rows x 32 floats = 2048 b128 transfers -> 8 per thread (coalesced).
#pragma unroll
  for (int j = 0; j < 8; ++j) {
    const int i  = tid + j * 256;
    const int r  = i >> 3;          // 8 float4 per row
    const int c4 = i & 7;
    async_copy_b128(aBase + (uint32_t)(r * AST + c4 * 4) * 4u,
                    xq + (rowBase + r) * (long)D + kc + c4 * 4);
  }
  // B: 64 rows x 32 floats = 512 b128 transfers -> 2 per thread (L2-resident).
#pragma unroll
  for (int j = 0; j < 2; ++j) {
    const int i  = tid + j * 256;
    const int n  = i >> 3;
    const int c4 = i & 7;
    async_copy_b128(bBase + (uint32_t)(n * BST + c4 * 4) * 4u,
                    proto + n * D + kc + c4 * 4);
  }
}

__global__ __launch_bounds__(256) void proto_nca_loss_kernel(
    const float* __restrict__ xq,     // [M, D]
    const float* __restrict__ proto,  // [KCLS, D]
    float* __restrict__ out) {        // out[0]=loss, out[1]=acc
  __shared__ float smem[SMEM_FLOATS];

  const int tid  = threadIdx.x;
  const int wave = tid >> 5;
  const int lane = tid & 31;
  const int l16  = lane & 15;
  const int hi   = lane >> 4;                 // selects K+0/+1 vs K+2/+3 lane half
  const long rowBase = (long)blockIdx.x * ROWS;

  v8f acc0[4], acc1[4];
#pragma unroll
  for (int i = 0; i < 4; ++i) {
    acc0[i] = (v8f){0.f, 0.f, 0.f, 0.f, 0.f, 0.f, 0.f, 0.f};
    acc1[i] = (v8f){0.f, 0.f, 0.f, 0.f, 0.f, 0.f, 0.f, 0.f};
  }
  float prsqAcc = 0.0f;

  // Software pipeline: prefetch chunk 0, then {wait, prefetch next, compute cur}.
  issue_chunk_loads(smem, 0, xq, proto, rowBase, 0, tid);

  int cur = 0;
#pragma unroll 1
  for (int ci = 0; ci < NCHUNKS; ++ci) {
    wait_async_all();      // this wave's async writes to smem are complete
    __syncthreads();       // -> every wave's writes complete; prev readers done
    if (ci + 1 < NCHUNKS)
      issue_chunk_loads(smem, cur ^ 1, xq, proto, rowBase, (ci + 1) * KC, tid);

    const int aOffF = cur ? OFF_A1 : OFF_A0;
    const int bOffF = cur ? OFF_B1 : OFF_B0;

    // pr_sq accumulated for free from the staged proto chunk.
    if (tid < KCLS) {
#pragma unroll 4
      for (int kk = 0; kk < KC; ++kk) {
        const float v = smem[bOffF + tid * BST + kk];
        prsqAcc = fmaf(v, v, prsqAcc);
      }
    }

    // Each wave: two 16x64 C tiles; B fragments shared by both tiles.
    // f32 16x16x4 fragment layout (ISA 7.12.2): VGPR0 lanes0-15 K+0 / lanes16-31 K+2,
    // VGPR1 K+1 / K+3; A indexed by M=lane%16, B by N=lane%16. Row-major ldsB[n][kk]
    // makes each B fragment 2 contiguous floats -> one conflict-free ds_load_b64.
    const float* aP0 = &smem[aOffF + (wave * 16 + l16) * AST];
    const float* aP1 = aP0 + 128 * AST;
    const float* bP  = &smem[bOffF + l16 * BST];
#pragma unroll
    for (int kk = 0; kk < KC; kk += 4) {
      const int ka = kk + 2 * hi;
      const v2f a0 = *reinterpret_cast<const v2f*>(aP0 + ka);
      const v2f a1 = *reinterpret_cast<const v2f*>(aP1 + ka);
      v2f b[4];
#pragma unroll
      for (int nt = 0; nt < 4; ++nt)
        b[nt] = *reinterpret_cast<const v2f*>(bP + nt * 16 * BST + ka);
#pragma unroll
      for (int nt = 0; nt < 4; ++nt) {
        acc0[nt] = __builtin_amdgcn_wmma_f32_16x16x4_f32(
            false, a0, false, b[nt], (short)0, acc0[nt], false, false);
        acc1[nt] = __builtin_amdgcn_wmma_f32_16x16x4_f32(
            false, a1, false, b[nt], (short)0, acc1[nt], false, false);
      }
    }
    cur ^= 1;
  }

  if (tid < KCLS) smem[OFF_P + tid] = prsqAcc;
  __syncthreads();  // all GEMM reads done; safe to overwrite A buffers with C

  // Spill C (ISA 7.12.2: VGPR v -> M=v for lanes0-15, M=8+v for lanes16-31).
#pragma unroll
  for (int nt = 0; nt < 4; ++nt)
#pragma unroll
    for (int v = 0; v < 8; ++v) {
      const int mrow = wave * 16 + hi * 8 + v;
      smem[mrow * CST + nt * 16 + l16]         = acc0[nt][v];
      smem[(128 + mrow) * CST + nt * 16 + l16] = acc1[nt][v];
    }
  __syncthreads();

  // One row per thread. xq_sq cancels in log_softmax: s_j = 2*cross_j - pr_sq_j.
  const long gRow  = rowBase + tid;
  const int  label = (int)(gRow >> 9);   // / QPC
  float* c = &smem[tid * CST];
  float mx = -3.402823466e38f;
  int pred = 0;
#pragma unroll 4
  for (int j = 0; j < KCLS; ++j) {
    const float s = 2.0f * c[j] - smem[OFF_P + j];
    c[j] = s;
    if (s > mx) { mx = s; pred = j; }    // strict > == first max, matches argmax
  }
  float sum = 0.0f;
#pragma unroll 4
  for (int j = 0; j < KCLS; ++j) sum += __expf(c[j] - mx);
  const float lse = mx + __logf(sum);
  float lossRow = 0.5f * (lse - c[label]);
  float hit     = (pred == label) ? 1.0f : 0.0f;

  // Wave reduction, then one pair of global f32 atomics per wave.
#pragma unroll
  for (int off = 16; off > 0; off >>= 1) {
    lossRow += __shfl_down(lossRow, off, 32);
    hit     += __shfl_down(hit, off, 32);
  }
  if (lane == 0) {
    atomicAdd(&out[0], lossRow * (1.0f / (float)M));
    atomicAdd(&out[1], hit * (1.0f / (float)M));
  }
}

extern "C" void kernel_launch(void* const* d_in, const int* in_sizes, int n_in,
                              void* d_out, int out_size, void* d_ws, size_t ws_size,
                              hipStream_t stream) {
  const float* xq    = (const float*)d_in[0];   // [64, 512, 1024] f32
  const float* proto = (const float*)d_in[1];   // [64, 1024] f32
  float* out = (float*)d_out;                   // [loss, acc]
  (void)in_sizes; (void)n_in; (void)out_size; (void)d_ws; (void)ws_size;

  zero_out_kernel<<<1, 32, 0, stream>>>(out);
  proto_nca_loss_kernel<<<M / ROWS, 256, 0, stream>>>(xq, proto, out);
}